// Counter_52879637348457
// MI455X (gfx1250) — compile-verified
//
#include <hip/hip_runtime.h>
#include <hip/hip_bf16.h>
#include <stdint.h>

#define M   64
#define NPW 16

typedef __attribute__((ext_vector_type(2))) float v2f;
typedef __attribute__((ext_vector_type(8))) float v8f;

// Exact replica of reference plin on normalized tables (cs = cumsum(w), w normalized).
__device__ __forceinline__ float plinf(const float* cs, const float* w, float x) {
    float y   = 16.0f * x;
    int   idx = (int)y;                       // trunc (matches astype(int32))
    idx = idx < 0 ? 0 : (idx > NPW ? NPW : idx);
    int   i1  = idx + 1; if (i1 > NPW) i1 = NPW;
    return cs[idx] + (y - floorf(y)) * w[i1];
}

__global__ __launch_bounds__(256)
void counter_score_kernel(const float* __restrict__ g_boxes,
                          const float* __restrict__ g_att,
                          const float* __restrict__ g_fw,
                          float* __restrict__ g_out) {
    __shared__ __attribute__((aligned(16))) float sm_boxes[4 * M];   // [coord][j]
    __shared__ __attribute__((aligned(16))) float sm_att[M];
    __shared__ float sm_area[M];
    __shared__ float sm_w[8][NPW + 1];
    __shared__ float sm_csum[8][NPW + 1];
    __shared__ float sm_dist[M * M];   // dist[j][k]
    __shared__ float sm_ds[M * M];     // dedup_score[i][j]
    __shared__ float sm_sim[M * M];    // sim[j][k]
    __shared__ float sm_red[256];
    __shared__ float sm_invr[M];
    __shared__ float sm_scalars[4];    // 0:dist_conf_sum 1:corr_sum 2:att_conf_sum

    const int tid = threadIdx.x;
    const int b   = blockIdx.x;

    // ---- Stage 0: async-copy boxes (1KB) + attention (256B) into LDS (CDNA5 path)
    if (tid < 80) {
        const float* gp;
        unsigned     loff;
        if (tid < 64) {
            gp   = g_boxes + (size_t)b * (4 * M) + tid * 4;
            loff = (unsigned)(size_t)&sm_boxes[tid * 4];
        } else {
            int t2 = tid - 64;
            gp   = g_att + (size_t)b * M + t2 * 4;
            loff = (unsigned)(size_t)&sm_att[t2 * 4];
        }
        asm volatile("global_load_async_to_lds_b128 %0, %1, off"
                     :: "v"(loff), "v"(gp) : "memory");
    }
    // Prepare normalized piecewise-linear tables for f0..f7 (overlaps the async copy)
    if (tid < 8) {
        float wv[NPW + 1];
        float s = 0.0f;
#pragma unroll
        for (int i = 0; i <= NPW; ++i) { wv[i] = fabsf(g_fw[tid * (NPW + 1) + i]); s += wv[i]; }
        float inv = 1.0f / s, c = 0.0f;
#pragma unroll
        for (int i = 0; i <= NPW; ++i) {
            float t = wv[i] * inv;
            c += t;
            sm_w[tid][i]    = t;
            sm_csum[tid][i] = c;
        }
    }
    asm volatile("s_wait_asynccnt 0x0" ::: "memory");
    __syncthreads();

    // ---- Stage 1: box areas
    if (tid < M) {
        int   j  = tid;
        float dx = fmaxf(sm_boxes[2 * M + j] - sm_boxes[0 * M + j], 0.0f);
        float dy = fmaxf(sm_boxes[3 * M + j] - sm_boxes[1 * M + j], 0.0f);
        sm_area[j] = dx * dy;
    }
    __syncthreads();

    // ---- Stage 2: dist = 1 - IoU, ds = f3(rel)*f4(dist), dist_conf partials
    {
        const float* cs3 = sm_csum[3]; const float* w3 = sm_w[3];
        const float* cs4 = sm_csum[4]; const float* w4 = sm_w[4];
        const float* cs6 = sm_csum[6]; const float* w6 = sm_w[6];
        float dconf = 0.0f;
        for (int e = tid; e < M * M; e += 256) {
            int   j = e >> 6, k = e & 63;
            float mnx = fmaxf(sm_boxes[j],         sm_boxes[k]);
            float mny = fmaxf(sm_boxes[M + j],     sm_boxes[M + k]);
            float mxx = fminf(sm_boxes[2 * M + j], sm_boxes[2 * M + k]);
            float mxy = fminf(sm_boxes[3 * M + j], sm_boxes[3 * M + k]);
            float iw = fmaxf(mxx - mnx, 0.0f);
            float ih = fmaxf(mxy - mny, 0.0f);
            float inter = iw * ih;
            float iouv  = inter / (sm_area[j] + sm_area[k] - inter + 1e-12f);
            float dist  = 1.0f - iouv;
            sm_dist[e]  = dist;
            float rel   = sm_att[j] * sm_att[k];
            sm_ds[e]    = plinf(cs3, w3, rel) * plinf(cs4, w4, dist);
            dconf      += fabsf(plinf(cs6, w6, dist) - 0.5f);
        }
        sm_red[tid] = dconf;
        __syncthreads();
        for (int s = 128; s > 0; s >>= 1) {
            if (tid < s) sm_red[tid] += sm_red[tid + s];
            __syncthreads();
        }
        if (tid == 0) sm_scalars[0] = sm_red[0];
        __syncthreads();
    }

    // ---- Stage 3: sim[j,k] = prod_i f2(1-|ds[i,j]-ds[i,k]|) * f2(1-|att_j-att_k|)
    {
        const float* cs2 = sm_csum[2]; const float* w2 = sm_w[2];
        for (int e = tid; e < M * M; e += 256) {
            int j = e >> 6, k = e & 63;
            const float* dsj = sm_ds + j;
            const float* dsk = sm_ds + k;
            float p = 1.0f;
#pragma unroll 4
            for (int i = 0; i < M; ++i) {
                float diff = fabsf(dsj[i * M] - dsk[i * M]);
                p *= plinf(cs2, w2, 1.0f - diff);
            }
            p *= plinf(cs2, w2, 1.0f - fabsf(sm_att[j] - sm_att[k]));
            sm_sim[e] = p;
        }
        __syncthreads();
    }

    // ---- Stage 4: row sums -> u = 1/r; correction & att_conf partials
    if (tid < M) {
        int   j = tid;
        float r = 0.0f;
        for (int k = 0; k < M; ++k) r += sm_sim[j * M + k];
        float ir   = 1.0f / r;
        sm_invr[j] = ir;
        float av   = sm_att[j];
        sm_red[j]      = plinf(sm_csum[0], sm_w[0], av * av) * ir;   // correction_j
        sm_red[64 + j] = fabsf(plinf(sm_csum[5], sm_w[5], av) - 0.5f); // att_conf_j
    }
    __syncthreads();
    if (tid == 0) {
        float c = 0.0f, a = 0.0f;
        for (int j = 0; j < M; ++j) { c += sm_red[j]; a += sm_red[64 + j]; }
        sm_scalars[1] = c;
        sm_scalars[2] = a;
    }
    __syncthreads();

    // ---- Stage 5: bilinear form u^T A u with rank-1 WMMA outer products.
    // rel = att (x) att and U = u (x) u computed by V_WMMA_F32_16X16X4_F32;
    // the WMMA C-layout (lane L, vgpr v -> row M = v + 8*(L>>4), col N = L&15)
    // IS the element decomposition for the f0/f1 elementwise stage.
    {
        const float* cs0 = sm_csum[0]; const float* w0 = sm_w[0];
        const float* cs1 = sm_csum[1]; const float* w1 = sm_w[1];
        const int lane = tid & 31;
        const int wave = tid >> 5;
        float acc = 0.0f;
#pragma unroll
        for (int it = 0; it < 2; ++it) {
            int t  = wave * 2 + it;            // 16 tiles of 16x16 over (j,k)
            int j0 = (t >> 2) * 16;
            int k0 = (t & 3) * 16;
            // A-layout (16x4 f32): lanes0-15 M=lane, vgpr0={K0,K2}, vgpr1={K1,K3}.
            // Only K=0 column nonzero -> rank-1 outer product.
            float aj = (lane < 16) ? sm_att[j0 + lane]  : 0.0f;
            float bk = (lane < 16) ? sm_att[k0 + lane]  : 0.0f;
            float uj = (lane < 16) ? sm_invr[j0 + lane] : 0.0f;
            float uk = (lane < 16) ? sm_invr[k0 + lane] : 0.0f;
            v2f Aa = {aj, 0.0f}, Bb = {bk, 0.0f};
            v2f Ua = {uj, 0.0f}, Ub = {uk, 0.0f};
            v8f z  = {};
            v8f rel8 = __builtin_amdgcn_wmma_f32_16x16x4_f32(false, Aa, false, Bb,
                                                             (short)0, z, false, false);
            v8f u8   = __builtin_amdgcn_wmma_f32_16x16x4_f32(false, Ua, false, Ub,
                                                             (short)0, z, false, false);
            int kk    = k0 + (lane & 15);
            int jbase = j0 + ((lane >> 4) << 3);
#pragma unroll
            for (int v = 0; v < 8; ++v) {
                int   jj = jbase + v;
                float d  = sm_dist[jj * M + kk];
                acc += plinf(cs0, w0, rel8[v]) * plinf(cs1, w1, d) * u8[v];
            }
        }
        sm_red[tid] = acc;
        __syncthreads();
        for (int s = 128; s > 0; s >>= 1) {
            if (tid < s) sm_red[tid] += sm_red[tid + s];
            __syncthreads();
        }
    }

    // ---- Stage 6: score -> one_hot * conf
    if (tid == 0) {
        float score = sqrtf(sm_red[0] + sm_scalars[1] + 1e-20f);
        float conf  = plinf(sm_csum[7], sm_w[7],
                            sm_scalars[2] * (1.0f / 64.0f) +
                            sm_scalars[0] * (1.0f / 4096.0f));
        float sc = fminf(fmaxf(score, 0.0f), 10.0f);
        int   i  = (int)sc;
        if (i > 10) i = 10;
        float fr = sc - floorf(sc);
        int   i1 = (i + 1 > 10) ? 10 : i + 1;
#pragma unroll
        for (int s2 = 0; s2 < 11; ++s2) {
            float vv = (1.0f - fr) * ((s2 == i)  ? 1.0f : 0.0f)
                     +         fr  * ((s2 == i1) ? 1.0f : 0.0f);
            g_out[b * 11 + s2] = vv * conf;
        }
    }
}

extern "C" void kernel_launch(void* const* d_in, const int* in_sizes, int n_in,
                              void* d_out, int out_size, void* d_ws, size_t ws_size,
                              hipStream_t stream) {
    (void)n_in; (void)out_size; (void)d_ws; (void)ws_size;
    const float* boxes = (const float*)d_in[0];
    const float* att   = (const float*)d_in[1];
    const float* fw    = (const float*)d_in[2];
    float*       out   = (float*)d_out;
    int n = in_sizes[1] / M;   // attention is (n, 64)
    counter_score_kernel<<<n, 256, 0, stream>>>(boxes, att, fw, out);
}